// Seq2seqMwerLoss_24730421690615
// MI455X (gfx1250) — compile-verified
//
#include <hip/hip_runtime.h>
#include <stdint.h>

// Problem constants (from reference setup_inputs)
#define BB    16
#define SS    256
#define VV    16000
#define NROWS (BB * SS)
#define BEAM  4
#define T1    512           // threads per block in kernel 1 (16 waves)
#define NWAVE (T1 / 32)

#define NEG_INF (-__builtin_inff())

// ---------------------------------------------------------------------------
// small helpers: constant-index selects so 4-element "arrays" stay in VGPRs
// ---------------------------------------------------------------------------
__device__ __forceinline__ float self4(const float v[4], int i) {
    float r = v[0];
    r = (i == 1) ? v[1] : r;
    r = (i == 2) ? v[2] : r;
    r = (i == 3) ? v[3] : r;
    return r;
}
__device__ __forceinline__ int seli4(const int v[4], int i) {
    int r = v[0];
    r = (i == 1) ? v[1] : r;
    r = (i == 2) ? v[2] : r;
    r = (i == 3) ? v[3] : r;
    return r;
}

// merge two descending-sorted (value,index) top-4 lists into av/ai.
// Ties broken toward the smaller index (jax top_k convention).
__device__ __forceinline__ void merge4(float av[4], int ai[4],
                                       const float bv[4], const int bi[4]) {
    float rv[4]; int ri[4];
    int i = 0, j = 0;
#pragma unroll
    for (int k = 0; k < 4; ++k) {
        float a = self4(av, i), b = self4(bv, j);
        int   x = seli4(ai, i), y = seli4(bi, j);
        bool ta = (a > b) || (a == b && x <= y);
        rv[k] = ta ? a : b;
        ri[k] = ta ? x : y;
        i += ta ? 1 : 0;
        j += ta ? 0 : 1;
    }
#pragma unroll
    for (int k = 0; k < 4; ++k) { av[k] = rv[k]; ai[k] = ri[k]; }
}

// ---------------------------------------------------------------------------
// Kernel 1: per (b,t) row of V=16000 floats compute
//   lse  = logsumexp(row)
//   top4 values (raw logits, desc sorted) + their vocab indices
// Row staged in LDS once via CDNA5 async global->LDS B128 loads (NT hint:
// pure streaming, keep the 192MB L2 clean), then scanned twice from LDS
// (max+top4, then sum-exp). HBM read exactly once.
// ---------------------------------------------------------------------------
__global__ __launch_bounds__(T1)
void k_rowstats(const float* __restrict__ logit,
                float* __restrict__ o_lse,
                float* __restrict__ o_t4v,
                int*   __restrict__ o_t4i)
{
    extern __shared__ char smem[];
    float* s_row = (float*)smem;                         // VV floats (64000 B)
    float* s_red = (float*)(smem + VV * 4);              // NWAVE floats
    float* s_wv  = (float*)(smem + (VV + NWAVE) * 4);    // NWAVE * 4
    int*   s_wi  = (int*)  (smem + (VV + NWAVE + NWAVE * 4) * 4); // NWAVE * 4

    const int tid  = threadIdx.x;
    const int lane = tid & 31;
    const int wave = tid >> 5;
    const int row  = blockIdx.x;
    const float* rowp = logit + (size_t)row * VV;

    // ---- async copy: global -> LDS, 16B per lane per issue (ASYNCcnt path)
    uint32_t lds_base = (uint32_t)(uintptr_t)s_row;      // low 32b of flat = LDS offset
    for (int c = tid; c < VV / 4; c += T1) {
        uint32_t lo = lds_base + (uint32_t)(c * 16);
        uint32_t go = (uint32_t)(c * 16);
        asm volatile("global_load_async_to_lds_b128 %0, %1, %2 th:TH_LOAD_NT"
                     :: "v"(lo), "v"(go), "s"(rowp) : "memory");
    }
    asm volatile("s_wait_asynccnt 0" ::: "memory");
    __syncthreads();

    // ---- pass A: per-thread top-4 over strided LDS scan (bank-conflict free)
    float tv[4] = {NEG_INF, NEG_INF, NEG_INF, NEG_INF};
    int   ti[4] = {0x7fffffff, 0x7fffffff, 0x7fffffff, 0x7fffffff};
    for (int t = tid; t < VV; t += T1) {
        float x = s_row[t];
        if (x > tv[3]) {
            tv[3] = x; ti[3] = t;
#pragma unroll
            for (int k = 3; k > 0; --k) {
                if (tv[k] > tv[k - 1]) {
                    float fv = tv[k]; tv[k] = tv[k - 1]; tv[k - 1] = fv;
                    int   fi = ti[k]; ti[k] = ti[k - 1]; ti[k - 1] = fi;
                }
            }
        }
    }

    // ---- wave32 shuffle-merge (down-reduce to lane 0 of each wave)
#pragma unroll
    for (int off = 16; off >= 1; off >>= 1) {
        float bv[4]; int bi[4];
#pragma unroll
        for (int k = 0; k < 4; ++k) {
            bv[k] = __shfl_down(tv[k], off, 32);
            bi[k] = __shfl_down(ti[k], off, 32);
        }
        merge4(tv, ti, bv, bi);
    }
    if (lane == 0) {
#pragma unroll
        for (int k = 0; k < 4; ++k) { s_wv[wave * 4 + k] = tv[k]; s_wi[wave * 4 + k] = ti[k]; }
    }
    __syncthreads();

    // ---- wave 0 merges the NWAVE wave-level lists
    if (tid < 32) {
        float cv[4]; int ci[4];
        if (tid < NWAVE) {
#pragma unroll
            for (int k = 0; k < 4; ++k) { cv[k] = s_wv[tid * 4 + k]; ci[k] = s_wi[tid * 4 + k]; }
        } else {
#pragma unroll
            for (int k = 0; k < 4; ++k) { cv[k] = NEG_INF; ci[k] = 0x7fffffff; }
        }
#pragma unroll
        for (int off = 8; off >= 1; off >>= 1) {
            float bv[4]; int bi[4];
#pragma unroll
            for (int k = 0; k < 4; ++k) {
                bv[k] = __shfl_down(cv[k], off, 32);
                bi[k] = __shfl_down(ci[k], off, 32);
            }
            merge4(cv, ci, bv, bi);
        }
        if (tid == 0) {
#pragma unroll
            for (int k = 0; k < 4; ++k) { s_wv[k] = cv[k]; s_wi[k] = ci[k]; }
        }
    }
    __syncthreads();

    // ---- pass B: sum exp(x - m) from LDS, wave shuffle reductions
    const float m = s_wv[0];
    float p = 0.f;
    for (int t = tid; t < VV; t += T1) p += __expf(s_row[t] - m);
#pragma unroll
    for (int off = 16; off >= 1; off >>= 1) p += __shfl_down(p, off, 32);
    if (lane == 0) s_red[wave] = p;
    __syncthreads();

    if (tid < 32) {
        float q = (tid < NWAVE) ? s_red[tid] : 0.f;
#pragma unroll
        for (int off = 8; off >= 1; off >>= 1) q += __shfl_down(q, off, 32);
        if (tid == 0) {
            o_lse[row] = m + __logf(q);
#pragma unroll
            for (int k = 0; k < 4; ++k) {
                o_t4v[(size_t)row * 4 + k] = s_wv[k];
                o_t4i[(size_t)row * 4 + k] = s_wi[k];
            }
        }
    }
}

// ---------------------------------------------------------------------------
// Kernel 2: one block per batch. Preload batch's top-4/LSE data into LDS,
// then a register-resident sequential beam scan over t < len, accumulating
// WER error counts and gathered log-prob sums on the fly, then the per-batch
// MWER loss. (Steps t >= len provably don't affect the output.)
// ---------------------------------------------------------------------------
__global__ __launch_bounds__(256)
void k_beam(const float* __restrict__ ws_lse,
            const float* __restrict__ ws_t4v,
            const int*   __restrict__ ws_t4i,
            const int*   __restrict__ tgt,
            const int*   __restrict__ tgt_lens,
            float* __restrict__ o_loss)
{
    __shared__ float ls_s[SS * 4];   // log_post value of top-4 per step
    __shared__ int   ls_v[SS * 4];   // vocab index of top-4 per step
    __shared__ int   ls_t[SS];       // targets

    const int b = blockIdx.x;
    const int tid = threadIdx.x;
    int len = tgt_lens[b];
    len = len < 1 ? 1 : (len > SS ? SS : len);

    for (int k = tid; k < SS * 4; k += 256) {
        ls_s[k] = ws_t4v[(size_t)b * SS * 4 + k] - ws_lse[b * SS + (k >> 2)];
        ls_v[k] = ws_t4i[(size_t)b * SS * 4 + k];
    }
    for (int k = tid; k < SS; k += 256) ls_t[k] = tgt[b * SS + k];
    __syncthreads();

    if (tid == 0) {
        float lp[4], gsum[4];
        int   err[4];
        // step t = 0: top-4 of log_post directly (mask[0] is always false)
#pragma unroll
        for (int j = 0; j < 4; ++j) {
            lp[j]   = ls_s[j];
            gsum[j] = ls_s[j];
            err[j]  = (ls_v[j] != ls_t[0]) ? 1 : 0;
        }
        for (int t = 1; t < len; ++t) {
            float sj[4]; int vj[4];
#pragma unroll
            for (int j = 0; j < 4; ++j) { sj[j] = ls_s[t * 4 + j]; vj[j] = ls_v[t * 4 + j]; }
            float c[16]; int fl[16];
#pragma unroll
            for (int i = 0; i < 4; ++i)
#pragma unroll
                for (int j = 0; j < 4; ++j) {
                    c[i * 4 + j]  = lp[i] + sj[j];
                    fl[i * 4 + j] = i * VV + vj[j];   // flat index for tie-break
                }
            float nlp[4];
#pragma unroll
            for (int r = 0; r < 4; ++r) {
                float bv = c[0]; int bf = fl[0]; int bk = 0;
#pragma unroll
                for (int k = 1; k < 16; ++k) {
                    bool tk = (c[k] > bv) || (c[k] == bv && fl[k] < bf);
                    bv = tk ? c[k]  : bv;
                    bf = tk ? fl[k] : bf;
                    bk = tk ? k     : bk;
                }
                nlp[r] = bv;
                const int j = bk & 3;
                gsum[r] += self4(sj, j);
                err[r]  += (seli4(vj, j) != ls_t[t]) ? 1 : 0;
#pragma unroll
                for (int k = 0; k < 16; ++k)
                    if (k == bk) c[k] = NEG_INF;     // remove winner
            }
#pragma unroll
            for (int r = 0; r < 4; ++r) lp[r] = nlp[r];
        }

        // MWER loss for this batch: w = softmax_n(gsum), loss = sum_n w*(err - mean(err))
        float mg = fmaxf(fmaxf(gsum[0], gsum[1]), fmaxf(gsum[2], gsum[3]));
        float w[4], z = 0.f;
#pragma unroll
        for (int r = 0; r < 4; ++r) { w[r] = __expf(gsum[r] - mg); z += w[r]; }
        float emean = 0.25f * (float)(err[0] + err[1] + err[2] + err[3]);
        float lo = 0.f;
#pragma unroll
        for (int r = 0; r < 4; ++r) lo += (w[r] / z) * ((float)err[r] - emean);
        o_loss[b] = lo;
    }
}

// ---------------------------------------------------------------------------
// Kernel 3: mean over batches -> scalar output
// ---------------------------------------------------------------------------
__global__ void k_final(const float* __restrict__ o_loss, float* __restrict__ out)
{
    if (threadIdx.x == 0 && blockIdx.x == 0) {
        float s = 0.f;
#pragma unroll
        for (int b = 0; b < BB; ++b) s += o_loss[b];
        out[0] = s * (1.0f / (float)BB);
    }
}

// ---------------------------------------------------------------------------
// Launch
// ---------------------------------------------------------------------------
extern "C" void kernel_launch(void* const* d_in, const int* in_sizes, int n_in,
                              void* d_out, int out_size, void* d_ws, size_t ws_size,
                              hipStream_t stream)
{
    const float* logit = (const float*)d_in[0];
    const int*   tgt   = (const int*)d_in[1];
    const int*   lens  = (const int*)d_in[2];
    float*       out   = (float*)d_out;

    // workspace layout (floats): lse[NROWS] | t4v[4*NROWS] | t4i[4*NROWS] | loss[BB]
    float* ws      = (float*)d_ws;
    float* ws_lse  = ws;
    float* ws_t4v  = ws + NROWS;
    int*   ws_t4i  = (int*)(ws + NROWS * 5);
    float* ws_loss = ws + NROWS * 9;

    // s_row + s_red + s_wv + s_wi
    const size_t smem1 = (size_t)(VV + NWAVE + NWAVE * 4 + NWAVE * 4) * 4;

    k_rowstats<<<NROWS, T1, smem1, stream>>>(logit, ws_lse, ws_t4v, ws_t4i);
    k_beam<<<BB, 256, 0, stream>>>(ws_lse, ws_t4v, ws_t4i, tgt, lens, ws_loss);
    k_final<<<1, 32, 0, stream>>>(ws_loss, out);
}